// AdditiveAttention_56513179681514
// MI455X (gfx1250) — compile-verified
//
#include <hip/hip_runtime.h>

// Additive attention (B=8, Lq=Lk=H=Dv=256) for gfx1250.
// Pipeline: [WMMA f32 projections] -> [fused tanh-score + mask, reg-blocked 4x4]
//           -> [softmax over QUERY axis (reference quirk)] -> [WMMA f32 attn@V]

typedef __attribute__((ext_vector_type(2))) float v2f;
typedef __attribute__((ext_vector_type(8))) float v8f;

#define HDIM   256
#define LQ     256
#define LK     256
#define BATCH  8
#define ROWS   (BATCH * LQ)          // 2048
#define MASKV  (-1000000.0f)

// ---------------- fast device math (TRANS pipe) ----------------
__device__ __forceinline__ float fast_tanh(float x) {
#if __has_builtin(__builtin_amdgcn_tanhf)
    return __builtin_amdgcn_tanhf(x);            // v_tanh_f32 (gfx1250 trans op)
#else
    // tanh(x) = 1 - 2/(e^{2x}+1); saturates correctly at +-1 for |x| large
    float t = __builtin_amdgcn_exp2f(x * 2.8853900817779268f); // e^{2x}
    return 1.0f - 2.0f * __builtin_amdgcn_rcpf(t + 1.0f);
#endif
}
__device__ __forceinline__ float fast_exp(float x) {
    return __builtin_amdgcn_exp2f(x * 1.4426950408889634f);
}

// ---------------- fp32 WMMA 16x16 tile GEMM ----------------
// D(16x16) = A(16xK) * B(Kx16), all fp32, one wave per tile.
// B_IS_NK=true : B stored as B[n][k] (nn.Linear weight layout)
// B_IS_NK=false: B stored as B[k][n] (values layout)
template <bool B_IS_NK>
__device__ __forceinline__ void wmma_tile_f32(const float* __restrict__ A, int lda,
                                              const float* __restrict__ Bm, int ldb,
                                              float* __restrict__ D, int ldd, int K) {
    const int lane = threadIdx.x & 31;
    const int half = lane >> 4;   // selects K pair (A/B) and M offset (D)
    const int l16  = lane & 15;
    v8f acc = {0.f, 0.f, 0.f, 0.f, 0.f, 0.f, 0.f, 0.f};
#if __has_builtin(__builtin_amdgcn_wmma_f32_16x16x4_f32)
    for (int k0 = 0; k0 < K; k0 += 4) {
        const int ka = k0 + (half << 1);
        // A 16x4: lanes 0-15 hold K=k0,k0+1; lanes 16-31 hold K=k0+2,k0+3
        v2f a = *(const v2f*)(A + l16 * lda + ka);
        v2f b;
        if (B_IS_NK) {
            b = *(const v2f*)(Bm + l16 * ldb + ka);      // contiguous pair
        } else {
            b.x = Bm[(ka + 0) * ldb + l16];              // strided rows
            b.y = Bm[(ka + 1) * ldb + l16];
        }
        acc = __builtin_amdgcn_wmma_f32_16x16x4_f32(
            false, a, false, b, (short)0, acc, false, false);
    }
#else
    for (int r = 0; r < 8; ++r) {
        float s = 0.0f;
        const int m = r + (half << 3);
        for (int k = 0; k < K; ++k)
            s += A[m * lda + k] * (B_IS_NK ? Bm[l16 * ldb + k] : Bm[k * ldb + l16]);
        acc[r] = s;
    }
#endif
#pragma unroll
    for (int r = 0; r < 8; ++r)
        D[(r + (half << 3)) * ldd + l16] = acc[r];
}

// ---------------- kernel 1: q & k projections (x @ W^T) ----------------
__global__ void __launch_bounds__(256) proj_kernel(
    const float* __restrict__ queries, const float* __restrict__ keyes,
    const float* __restrict__ Wq, const float* __restrict__ Wk,
    float* __restrict__ qproj, float* __restrict__ kproj) {
    const int gw    = blockIdx.x * 8 + (threadIdx.x >> 5);  // 0..4095
    const int which = gw >> 11;                             // 0=q, 1=k
    const int tt    = gw & 2047;                            // 128 x 16 tiles
    const int tm    = (tt >> 4) << 4;
    const int tn    = (tt & 15) << 4;
    const float* X = which ? keyes : queries;
    const float* W = which ? Wk : Wq;
    float*       O = which ? kproj : qproj;
    wmma_tile_f32<true>(X + tm * HDIM, HDIM, W + tn * HDIM, HDIM,
                        O + tm * HDIM + tn, HDIM, HDIM);
}

// ---------------- kernel 2: fused additive scores + mask ----------------
// Block = (b, 16 i-rows) x 256 j. Thread = 4 i x 4 j register tile.
// LDS k-tile stored transposed [hh][j] (pitch 260 -> 16B-aligned b128 reads).
// Per hh per thread: 1x ds_b128 (k) + 4x broadcast (q) + 1x broadcast (w)
// feeding 16 add + 16 v_tanh (TRANS) + 16 fma  => TRANS/VALU-bound.
#define CH      32
#define KPITCH  260   // multiple of 4 (16B rows), 260 % 64 == 4 (bank spread)
__global__ void __launch_bounds__(256) scores_kernel(
    const float* __restrict__ qproj, const float* __restrict__ kproj,
    const float* __restrict__ Wv, const int* __restrict__ valid_lens,
    float* __restrict__ scores) {
    __shared__ float sK[CH * KPITCH];   // [hh][j], j = 0..255
    __shared__ float sQ[16 * CH];       // [ii][hh], 16 i-rows per block
    __shared__ float sW[CH];

    const int t   = threadIdx.x;
    const int b   = blockIdx.x >> 4;          // 16 blocks per batch
    const int i0  = (blockIdx.x & 15) << 4;   // first of 16 i-rows
    const int j0  = (t & 63) << 2;            // 4 j-cols per thread
    const int iq0 = (t >> 6) << 2;            // 4 i-rows per thread (wave-uniform)

    const float* kbase = kproj + (size_t)b * LK * HDIM;
    const float* qbase = qproj + ((size_t)b * LQ + i0) * HDIM;

    float acc[4][4];
#pragma unroll
    for (int ii = 0; ii < 4; ++ii)
#pragma unroll
        for (int jj = 0; jj < 4; ++jj) acc[ii][jj] = 0.0f;

    for (int h0 = 0; h0 < HDIM; h0 += CH) {
        // stage k[b, 0:256, h0:h0+CH) transposed; coalesced 128B global reads
        for (int e = t; e < LK * CH; e += 256) {
            const int j  = e >> 5;            // CH == 32
            const int hh = e & (CH - 1);
            sK[hh * KPITCH + j] = kbase[j * HDIM + h0 + hh];
        }
        // stage 16 q rows x CH and the Wv chunk
        for (int e = t; e < 16 * CH; e += 256) {
            const int ii = e >> 5;
            const int hh = e & (CH - 1);
            sQ[ii * CH + hh] = qbase[ii * HDIM + h0 + hh];
        }
        if (t < CH) sW[t] = Wv[h0 + t];
        __syncthreads();

#pragma unroll 4
        for (int hh = 0; hh < CH; ++hh) {
            const float4 kv = *(const float4*)(sK + hh * KPITCH + j0); // b128
            const float  wv = sW[hh];
#pragma unroll
            for (int ii = 0; ii < 4; ++ii) {
                const float qv = sQ[(iq0 + ii) * CH + hh];             // broadcast
                acc[ii][0] += wv * fast_tanh(qv + kv.x);
                acc[ii][1] += wv * fast_tanh(qv + kv.y);
                acc[ii][2] += wv * fast_tanh(qv + kv.z);
                acc[ii][3] += wv * fast_tanh(qv + kv.w);
            }
        }
        __syncthreads();
    }

    const int vl = valid_lens[b];
#pragma unroll
    for (int ii = 0; ii < 4; ++ii) {
        float4 o;
        o.x = (j0 + 0 < vl) ? acc[ii][0] : MASKV;
        o.y = (j0 + 1 < vl) ? acc[ii][1] : MASKV;
        o.z = (j0 + 2 < vl) ? acc[ii][2] : MASKV;
        o.w = (j0 + 3 < vl) ? acc[ii][3] : MASKV;
        *(float4*)(scores + ((size_t)b * LQ + i0 + iq0 + ii) * LK + j0) = o; // b128
    }
}

// ---------------- kernel 3: softmax over the QUERY axis (axis=1) ----------------
__global__ void __launch_bounds__(256) softmax_q_kernel(float* __restrict__ scores) {
    const int b = blockIdx.x;
    const int j = threadIdx.x;
    float* base = scores + (size_t)b * LQ * LK + j;
    float m = -3.4e38f;
    for (int i = 0; i < LQ; ++i) m = fmaxf(m, base[i * LK]);
    float s = 0.0f;
    for (int i = 0; i < LQ; ++i) s += fast_exp(base[i * LK] - m);
    const float inv = __builtin_amdgcn_rcpf(s);
    for (int i = 0; i < LQ; ++i) base[i * LK] = fast_exp(base[i * LK] - m) * inv;
}

// ---------------- kernel 4: out = attn @ values ----------------
__global__ void __launch_bounds__(256) out_kernel(
    const float* __restrict__ attn, const float* __restrict__ values,
    float* __restrict__ out) {
    const int gw = blockIdx.x * 8 + (threadIdx.x >> 5);  // 0..2047
    const int b  = gw >> 8;
    const int tt = gw & 255;                             // 16 x 16 tiles
    const int tm = (tt >> 4) << 4;
    const int tn = (tt & 15) << 4;
    const float* A  = attn   + (size_t)b * LQ * LK + tm * LK;
    const float* Bp = values + (size_t)b * LK * HDIM + tn;
    float*       D  = out    + (size_t)b * LQ * HDIM + tm * HDIM + tn;
    wmma_tile_f32<false>(A, LK, Bp, HDIM, D, HDIM, LK);
}

// ---------------- host side ----------------
extern "C" void kernel_launch(void* const* d_in, const int* in_sizes, int n_in,
                              void* d_out, int out_size, void* d_ws, size_t ws_size,
                              hipStream_t stream) {
    const float* queries = (const float*)d_in[0];
    const float* keyes   = (const float*)d_in[1];
    const float* values  = (const float*)d_in[2];
    const int*   valid   = (const int*)d_in[3];
    const float* Wq      = (const float*)d_in[4];
    const float* Wk      = (const float*)d_in[5];
    const float* Wv      = (const float*)d_in[6];
    float* out = (float*)d_out;

    float* qproj  = (float*)d_ws;               // 2 MB
    float* kproj  = qproj + ROWS * HDIM;        // 2 MB
    float* scores = kproj + ROWS * HDIM;        // 2 MB, reused in-place as attn

    proj_kernel     <<<512, 256, 0, stream>>>(queries, keyes, Wq, Wk, qproj, kproj);
    scores_kernel   <<<128, 256, 0, stream>>>(qproj, kproj, Wv, valid, scores);
    softmax_q_kernel<<<BATCH, 256, 0, stream>>>(scores);
    out_kernel      <<<256, 256, 0, stream>>>(scores, values, out);
}